// KVCache_10986526343916
// MI455X (gfx1250) — compile-verified
//
#include <hip/hip_runtime.h>

// ---------------------------------------------------------------------------
// KV-cache roll for B=4, H=32, L=4096, D=128, S=1 (fp32).
//   out[b,h,l,:]    = cache[b,h,l+1,:]   (l < L-1)
//   out[b,h,L-1,:]  = new[b,h,0,:]
// Flat view: per (b,h) block of L*D floats, out[j] = cache[j+D] in the body,
// last D floats come from the new token row. Pure 1 GiB stream -> HBM bound
// (~46us @ 23.3 TB/s). Uses gfx1250 async global<->LDS B128 DMA with NT
// cache policy and double-buffered LDS so loads of iter i overlap stores of
// iter i-1 on the same ASYNCcnt.
// ---------------------------------------------------------------------------

typedef float __attribute__((ext_vector_type(4))) f4;
typedef int v4i __attribute__((vector_size(16)));   // matches builtin param type

#define AS1 __attribute__((address_space(1)))
#define AS3 __attribute__((address_space(3)))

static constexpr unsigned kBH   = 4u * 32u;          // B*H blocks
static constexpr unsigned kL    = 4096u;
static constexpr unsigned kD    = 128u;
static constexpr unsigned kLD   = kL * kD;           // 524288 = 2^19 floats/block
static constexpr unsigned kBODY = kLD - kD;          // body/new-row threshold
static constexpr unsigned kN    = kBH * kLD;         // 67,108,864 floats per tensor
static constexpr unsigned kNF4  = kN / 4u;           // 16,777,216 float4 per tensor

// CPol temporal hint: TH[2:0]=1 -> non-temporal (stream never revisits L2).
#define KV_CPOL_NT 1

#if __has_builtin(__builtin_amdgcn_global_load_async_to_lds_b128) && \
    __has_builtin(__builtin_amdgcn_global_store_async_from_lds_b128)
#define KV_USE_ASYNC_LDS 1
#else
#define KV_USE_ASYNC_LDS 0
#endif

__device__ __forceinline__ void kv_wait_async0() {
#if __has_builtin(__builtin_amdgcn_s_wait_asynccnt)
  __builtin_amdgcn_s_wait_asynccnt(0);
#else
  asm volatile("s_wait_asynccnt 0" ::: "memory");
#endif
}

// Per-lane source pointer: body comes from the shifted cache, the final row of
// each (b,h) block comes from the freshly appended token.
__device__ __forceinline__ const float* kv_src(const float* __restrict__ cache,
                                               const float* __restrict__ fresh,
                                               unsigned j, unsigned blk, unsigned r) {
  if (r < kBODY) return cache + (j + kD);
  return fresh + ((blk << 7) + (r - kBODY));   // blk*D + d
}

#if KV_USE_ASYNC_LDS
// DMA path: global -> LDS -> global, 16B per lane per tensor per step.
// ASYNCcnt discipline (per wave), double-buffered on (iteration & 1):
//   iter i: issue L(i) into buf[i&1]      (S(i-1) may still be in flight,
//                                          it reads buf[(i-1)&1] -> no clash)
//           s_wait_asynccnt 0             (drains L(i) AND S(i-1))
//           issue S(i) from buf[i&1]
// Buffer reuse is safe: when L(i) overwrites buf[i&1], its previous reader
// S(i-2) was drained by iteration (i-1)'s wait. S_ENDPGM waits idle at end.
__global__ void __launch_bounds__(256)
kv_roll_async(const float* __restrict__ kc, const float* __restrict__ vc,
              const float* __restrict__ kn, const float* __restrict__ vn,
              float* __restrict__ ok, float* __restrict__ ov) {
  __shared__ f4 bufk[2][256];
  __shared__ f4 bufv[2][256];
  const unsigned tid = threadIdx.x;
  AS3 v4i* lk0 = (AS3 v4i*)&bufk[0][tid];
  AS3 v4i* lk1 = (AS3 v4i*)&bufk[1][tid];
  AS3 v4i* lv0 = (AS3 v4i*)&bufv[0][tid];
  AS3 v4i* lv1 = (AS3 v4i*)&bufv[1][tid];

  const unsigned stride = gridDim.x * blockDim.x;          // in float4 units
  unsigned it = 0;
  for (unsigned i4 = blockIdx.x * blockDim.x + tid; i4 < kNF4; i4 += stride, ++it) {
    const unsigned j   = i4 << 2;          // float index
    const unsigned blk = j >> 19;          // j / (L*D)
    const unsigned r   = j & (kLD - 1u);   // j % (L*D)
    AS1 v4i* sk = (AS1 v4i*)kv_src(kc, kn, j, blk, r);
    AS1 v4i* sv = (AS1 v4i*)kv_src(vc, vn, j, blk, r);
    AS3 v4i* lk = (it & 1u) ? lk1 : lk0;
    AS3 v4i* lv = (it & 1u) ? lv1 : lv0;

    __builtin_amdgcn_global_load_async_to_lds_b128(sk, lk, 0, KV_CPOL_NT);
    __builtin_amdgcn_global_load_async_to_lds_b128(sv, lv, 0, KV_CPOL_NT);
    kv_wait_async0();  // loads landed; previous iteration's stores drained
    __builtin_amdgcn_global_store_async_from_lds_b128((AS1 v4i*)(ok + j), lk, 0, KV_CPOL_NT);
    __builtin_amdgcn_global_store_async_from_lds_b128((AS1 v4i*)(ov + j), lv, 0, KV_CPOL_NT);
  }
}
#else
// Fallback: plain 128-bit nontemporal copy (stream bypasses L2 retention).
__global__ void __launch_bounds__(256)
kv_roll_plain(const float* __restrict__ kc, const float* __restrict__ vc,
              const float* __restrict__ kn, const float* __restrict__ vn,
              float* __restrict__ ok, float* __restrict__ ov) {
  const unsigned stride = gridDim.x * blockDim.x;
  for (unsigned i4 = blockIdx.x * blockDim.x + threadIdx.x; i4 < kNF4; i4 += stride) {
    const unsigned j   = i4 << 2;
    const unsigned blk = j >> 19;
    const unsigned r   = j & (kLD - 1u);
    const f4 a = __builtin_nontemporal_load((const f4*)kv_src(kc, kn, j, blk, r));
    const f4 b = __builtin_nontemporal_load((const f4*)kv_src(vc, vn, j, blk, r));
    __builtin_nontemporal_store(a, (f4*)(ok + j));
    __builtin_nontemporal_store(b, (f4*)(ov + j));
  }
}
#endif

extern "C" void kernel_launch(void* const* d_in, const int* in_sizes, int n_in,
                              void* d_out, int out_size, void* d_ws, size_t ws_size,
                              hipStream_t stream) {
  const float* k_cache = (const float*)d_in[0];
  const float* v_cache = (const float*)d_in[1];
  const float* k_new   = (const float*)d_in[2];
  const float* v_new   = (const float*)d_in[3];
  // d_in[4] = context_length (== L), unused: shapes are static for this problem.

  float* out_k = (float*)d_out;
  float* out_v = out_k + kN;

  dim3 block(256);
  dim3 grid(4096);   // 1,048,576 threads -> 16 float4 per thread per tensor
#if KV_USE_ASYNC_LDS
  kv_roll_async<<<grid, block, 0, stream>>>(k_cache, v_cache, k_new, v_new, out_k, out_v);
#else
  kv_roll_plain<<<grid, block, 0, stream>>>(k_cache, v_cache, k_new, v_new, out_k, out_v);
#endif
}